// CholProdLayer_31748398252601
// MI455X (gfx1250) — compile-verified
//
#include <hip/hip_runtime.h>
#include <hip/hip_bf16.h>

typedef float v2f __attribute__((ext_vector_type(2)));
typedef float v8f __attribute__((ext_vector_type(8)));

#define KDIM   64
#define MDIM   512
#define NPROB  (KDIM * MDIM)        // 32768 independent 32x32 problems
#define DZ     528                  // 32*33/2
#define NMAT   32
#define LSTRIDE 36                  // row stride in floats: 8B-aligned rows, conflict-free
#define WAVES_PER_BLOCK 8

__global__ __launch_bounds__(256) void
cholprod_wmma_kernel(const float* __restrict__ z,
                     const float* __restrict__ sldj_in,
                     float* __restrict__ zout,
                     float* __restrict__ sldj_out)
{
    __shared__ float lds[WAVES_PER_BLOCK * NMAT * LSTRIDE];

    const int lane  = threadIdx.x & 31;
    const int wave  = threadIdx.x >> 5;
    const int prob  = blockIdx.x * WAVES_PER_BLOCK + wave;   // < 32768 always

    const int lr    = lane & 15;      // row/col within a 16-wide fragment
    const int khalf = lane >> 4;      // which half of the K/M split

    float* Lp = lds + wave * (NMAT * LSTRIDE);
    const float* zp = z + (size_t)prob * DZ;

    // ---- Build L_pos (diag exponentiated, upper tri zero) into LDS ----
    // fill_triangular inverse: t = 32*i + j (j<=i); src = (t<496) ? 32+t : 1023-t
    float dval = 0.0f;                 // lane j ends up holding L[j][j]
    #pragma unroll
    for (int i = 0; i < NMAT; ++i) {
        const int j = lane;
        float v = 0.0f;
        if (j <= i) {
            const int t   = i * NMAT + j;
            const int src = (t < 496) ? (32 + t) : (1023 - t);
            const float x = zp[src];
            if (j == i) { dval = x; v = __expf(x); }
            else        { v = x; }
        }
        Lp[i * LSTRIDE + j] = v;
    }
    __syncthreads();   // LDS visibility within the WGP (also reconverges EXEC for WMMA)

    // ---- C = L_pos * L_pos^T via V_WMMA_F32_16X16X4_F32 ----
    // A frag (16x4): lane -> row M = lr, VGPR pair holds K = 4*kk + 2*khalf + {0,1}
    // B frag (4x16) of B = L_pos^T: B[k][n] = L_pos[n][k] -> identical row-major read
    float* outp = zout + (size_t)prob * (NMAT * NMAT);

    #pragma unroll
    for (int ti = 0; ti < 2; ++ti) {
        const v2f* Arow = (const v2f*)(Lp + (ti * 16 + lr) * LSTRIDE);
        #pragma unroll
        for (int tj = 0; tj < 2; ++tj) {
            const v2f* Brow = (const v2f*)(Lp + (tj * 16 + lr) * LSTRIDE);
            v8f acc = {0.f, 0.f, 0.f, 0.f, 0.f, 0.f, 0.f, 0.f};
            #pragma unroll
            for (int kk = 0; kk < 8; ++kk) {
                v2f a = Arow[kk * 2 + khalf];   // K = 4*kk + 2*khalf .. +1
                v2f b = Brow[kk * 2 + khalf];
                acc = __builtin_amdgcn_wmma_f32_16x16x4_f32(
                    /*neg_a=*/false, a, /*neg_b=*/false, b,
                    /*c_mod=*/(short)0, acc,
                    /*reuse_a=*/false, /*reuse_b=*/false);
            }
            // D layout: VGPR r -> row M = ti*16 + khalf*8 + r, col N = tj*16 + lr
            #pragma unroll
            for (int r = 0; r < 8; ++r) {
                const int row = ti * 16 + khalf * 8 + r;
                const int col = tj * 16 + lr;
                float v = acc[r];
                if (row == col) v += 1e-6f;
                outp[row * NMAT + col] = v;     // coalesced 64B per half-wave
            }
        }
    }

    // ---- sldj: sldj_in + 32*ln2 + sum_j (33 - j) * L[j][j] ----
    float contrib = (float)(33 - lane) * dval;
    #pragma unroll
    for (int off = 16; off > 0; off >>= 1)
        contrib += __shfl_xor(contrib, off, 32);
    if (lane == 0)
        sldj_out[prob] = sldj_in[prob] + 22.180709777918249f /* 32*ln2 */ + contrib;
}

extern "C" void kernel_launch(void* const* d_in, const int* in_sizes, int n_in,
                              void* d_out, int out_size, void* d_ws, size_t ws_size,
                              hipStream_t stream)
{
    const float* z    = (const float*)d_in[0];      // [64,512,528,1] fp32
    const float* sldj = (const float*)d_in[1];      // [64,512] fp32
    float* out        = (float*)d_out;              // z_out then sldj, concatenated
    float* zout       = out;                        // 64*512*1024 floats
    float* sldj_out   = out + (size_t)NPROB * (NMAT * NMAT);

    (void)in_sizes; (void)n_in; (void)out_size; (void)d_ws; (void)ws_size;

    const int blocks = NPROB / WAVES_PER_BLOCK;     // 4096
    cholprod_wmma_kernel<<<blocks, 256, 0, stream>>>(z, sldj, zout, sldj_out);
}